// seg_loss_80633716015267
// MI455X (gfx1250) — compile-verified
//
#include <hip/hip_runtime.h>

#define NUM_CLASS 4
#define B_   16
#define H_   512
#define W_   512
#define PLANES (B_ * NUM_CLASS)     /* 64 */
#define TILE_H 16
#define BANDS  (H_ / TILE_H)        /* 32 */
#define NBLK   (PLANES * BANDS)     /* 2048 */
#define NTHREADS 256
#define SMOOTH 1e-4f
#define NTOT_F 16777216.0f          /* 16*4*512*512 */

#define PAD     4                   /* zero columns on each side (16B aligned) */
#define STRIDE  (W_ + 2 * PAD)      /* 520 ints per LDS row */
#define LROWS   (TILE_H + 2)        /* 18 rows incl. halo */

typedef __attribute__((ext_vector_type(2))) float v2f;
typedef __attribute__((ext_vector_type(4))) float v4f;
typedef __attribute__((ext_vector_type(8))) float v8f;
typedef __attribute__((ext_vector_type(4))) int   v4i;

// pointer-to-AS-qualified-v4i types for the async-to-LDS builtin
typedef __attribute__((address_space(1))) v4i* gas_v4i_ptr;  // global
typedef __attribute__((address_space(3))) v4i* las_v4i_ptr;  // LDS

#if defined(__HIP_DEVICE_COMPILE__) && __has_builtin(__builtin_amdgcn_wmma_f32_16x16x4_f32)
#define HAVE_WMMA_F32X4 1
#endif
#if defined(__HIP_DEVICE_COMPILE__) && __has_builtin(__builtin_amdgcn_global_load_async_to_lds_b128)
#define HAVE_ASYNC_LDS 1
#endif

__device__ __forceinline__ int imin(int a, int b) { return a < b ? a : b; }
__device__ __forceinline__ int imax(int a, int b) { return a > b ? a : b; }

// Full wave32 sum via V_WMMA_F32_16X16X4_F32 with a ones B-matrix.
// A (16x4 f32): lanes 0-15 vgpr0=K0, vgpr1=K1; lanes 16-31 vgpr0=K2, vgpr1=K3.
// With A vgpr0 = v, vgpr1 = 0:  D[m][n] = v[m] + v[m+16] for every n.
// Sum of the 8 C/D VGPRs per lane + xor-16 partner = full 32-lane total.
__device__ __forceinline__ float wave_sum(float v) {
#ifdef HAVE_WMMA_F32X4
  v2f a; a[0] = v;    a[1] = 0.0f;
  v2f b; b[0] = 1.0f; b[1] = 1.0f;
  v8f c = {0.f, 0.f, 0.f, 0.f, 0.f, 0.f, 0.f, 0.f};
  c = __builtin_amdgcn_wmma_f32_16x16x4_f32(false, a, false, b, (short)0, c,
                                            false, false);
  float s = c[0] + c[1] + c[2] + c[3] + c[4] + c[5] + c[6] + c[7];
  s += __shfl_xor(s, 16, 32);
  return s;
#else
  for (int off = 16; off > 0; off >>= 1) v += __shfl_xor(v, off, 32);
  return v;
#endif
}

__global__ __launch_bounds__(NTHREADS) void seg_loss_main(
    const float* __restrict__ cmap, const int* __restrict__ gt,
    float* __restrict__ ws) {
  __shared__ int s_gt[LROWS * STRIDE];      // ~36.6 KB padded band + halo
  __shared__ float s_part[8][4];

  const int bx    = blockIdx.x;
  const int plane = bx >> 5;                // 0..63  (b*C + c)
  const int band  = bx & 31;                // 0..31
  const int r0    = band * TILE_H;
  const size_t pbase  = (size_t)plane * (size_t)(H_ * W_);
  const int*   gplane = gt + pbase;
  const float* cplane = cmap + pbase;

  // ---- Zero the whole padded tile (halo rows/cols stay 0), then barrier.
  {
    const v4i z = {0, 0, 0, 0};
    for (int l4 = threadIdx.x; l4 < LROWS * (STRIDE / 4); l4 += NTHREADS)
      *(v4i*)&s_gt[l4 << 2] = z;
  }
  __syncthreads();

  // ---- Async-stage gt rows [r0-1, r0+TILE_H] into the interior.
  for (int l4 = threadIdx.x; l4 < LROWS * (W_ / 4); l4 += NTHREADS) {
    const int lrow = l4 >> 7;               // 128 int4 per data row
    const int lcol = (l4 & 127) << 2;       // data column in ints
    const int g = r0 - 1 + lrow;
    if (g >= 0 && g < H_) {
      const int* gp = gplane + g * W_ + lcol;
      int* lp = &s_gt[lrow * STRIDE + PAD + lcol];   // 16B aligned
#ifdef HAVE_ASYNC_LDS
      __builtin_amdgcn_global_load_async_to_lds_b128(
          (gas_v4i_ptr)gp, (las_v4i_ptr)lp, 0, 0);
#else
      *(int4*)lp = *(const int4*)gp;
#endif
    }
  }
#ifdef HAVE_ASYNC_LDS
#if __has_builtin(__builtin_amdgcn_s_wait_asynccnt)
  __builtin_amdgcn_s_wait_asynccnt(0);
#else
  asm volatile("s_wait_asynccnt 0" ::: "memory");
#endif
#endif
  __syncthreads();

  // ---- Branch-free streaming compute: 4 pixels per thread per iteration.
  float s_cee = 0.0f;   // sum of (softplus - x*y) + edge term
  float s_in  = 0.0f;   // sum y*pred
  float s_i   = 0.0f;   // sum y
  float s_j   = 0.0f;   // sum pred
  for (int it = 0; it < (TILE_H * W_) / (4 * NTHREADS); ++it) {  // 8 iters
    const int l4   = threadIdx.x + (it << 8);  // 0..2047 float4 slots
    const int row  = l4 >> 7;                  // 0..15
    const int col4 = (l4 & 127) << 2;          // 0,4,...,508
    const int gy   = r0 + row;

    const v4f xv = __builtin_nontemporal_load(
        reinterpret_cast<const v4f*>(cplane + gy * W_ + col4));

    const int base = (row + 1) * STRIDE + PAD + col4;
    const v4i c = *(const v4i*)&s_gt[base];            // center, 16B aligned
    const v4i u = *(const v4i*)&s_gt[base - STRIDE];   // up
    const v4i d = *(const v4i*)&s_gt[base + STRIDE];   // down
    const int lft = s_gt[base - 1];
    const int rgt = s_gt[base + 4];

    const int gls[4] = {lft, c[0], c[1], c[2]};
    const int grs[4] = {c[1], c[2], c[3], rgt};

#pragma unroll
    for (int j = 0; j < 4; ++j) {
      const int gc = c[j], gu = u[j], gd = d[j];
      const int dila = imax(imax(imax(imax(gc, gu), gd), gls[j]), grs[j]);
      const int eros = imin(imin(imin(imin(gc, gu), gd), gls[j]), grs[j]);
      const float edge = (float)dila * (1.0f - (float)eros);
      const float y = (float)gc;

      const float x = xv[j];
      const float e = __expf(-fabsf(x));
      const float r = __builtin_amdgcn_rcpf(1.0f + e);
      const float pred = (x >= 0.0f) ? r : e * r;          // sigmoid(x)
      const float sp = fmaxf(x, 0.0f) + __logf(1.0f + e);  // logaddexp(0,x)

      const float pe = pred * edge;
      s_cee += sp - x * y - fmaxf(__logf(1.0f - pe), -100.0f);
      s_in  += y * pred;
      s_i   += y;
      s_j   += pred;
    }
  }

  // ---- Reduction: WMMA ones-matmul per wave, then 8 waves via LDS.
  const float r_cee = wave_sum(s_cee);
  const float r_in  = wave_sum(s_in);
  const float r_i   = wave_sum(s_i);
  const float r_j   = wave_sum(s_j);
  const int wid = threadIdx.x >> 5;
  if ((threadIdx.x & 31) == 0) {
    s_part[wid][0] = r_cee; s_part[wid][1] = r_in;
    s_part[wid][2] = r_i;   s_part[wid][3] = r_j;
  }
  __syncthreads();
  if (threadIdx.x == 0) {
    float a0 = 0.f, a1 = 0.f, a2 = 0.f, a3 = 0.f;
    for (int w = 0; w < 8; ++w) {
      a0 += s_part[w][0]; a1 += s_part[w][1];
      a2 += s_part[w][2]; a3 += s_part[w][3];
    }
    float* o = ws + (size_t)bx * 8;
    o[0] = a0; o[1] = a1; o[2] = a2; o[3] = a3;
  }
}

__global__ __launch_bounds__(256) void seg_loss_finalize(
    const float* __restrict__ ws, float* __restrict__ out) {
  __shared__ float red[256];
  float loc = 0.0f;
  // ce_loss + edge_l: global mean of per-block combined sums
  for (int b = threadIdx.x; b < NBLK; b += 256) loc += ws[(size_t)b * 8 + 0];
  loc *= (1.0f / NTOT_F);
  // dice: one (b,c) plane per thread (first 64 threads)
  if ((int)threadIdx.x < PLANES) {
    const float* base = ws + (size_t)threadIdx.x * BANDS * 8;
    float in = 0.f, si = 0.f, sj = 0.f;
    for (int k = 0; k < BANDS; ++k) {
      in += base[k * 8 + 1];
      si += base[k * 8 + 2];
      sj += base[k * 8 + 3];
    }
    const float score = (2.0f * in + SMOOTH) / (si + sj + SMOOTH);
    loc += (1.0f - score) * (1.0f / (float)B_);   // mean over batch, sum over C
  }
  red[threadIdx.x] = loc;
  __syncthreads();
  for (int s = 128; s > 0; s >>= 1) {
    if ((int)threadIdx.x < s) red[threadIdx.x] += red[threadIdx.x + s];
    __syncthreads();
  }
  if (threadIdx.x == 0) out[0] = red[0];
}

extern "C" void kernel_launch(void* const* d_in, const int* in_sizes, int n_in,
                              void* d_out, int out_size, void* d_ws,
                              size_t ws_size, hipStream_t stream) {
  (void)in_sizes; (void)n_in; (void)out_size; (void)ws_size;
  const float* cmap = (const float*)d_in[0];
  const int*   gtp  = (const int*)d_in[1];
  float* out = (float*)d_out;
  float* ws  = (float*)d_ws;   // needs NBLK*8*4 = 64 KB

  seg_loss_main<<<NBLK, NTHREADS, 0, stream>>>(cmap, gtp, ws);
  seg_loss_finalize<<<1, 256, 0, stream>>>(ws, out);
}